// RecurrentLayer_27754078667490
// MI455X (gfx1250) — compile-verified
//
#include <hip/hip_runtime.h>

// uRNN (unitary RNN + modReLU) for MI455X / gfx1250.
// B=256,T=1024,I=64,H=256,O=64. fp32 WMMA (16x16x4) throughout.
// Pass 1: repack weights into WMMA-fragment-major planes in d_ws so every
//         B fragment is one contiguous global_load_b64 (no deinterleave movs
//         on the serial critical path).
// Pass 2: 16 persistent workgroups (one per 16-row batch tile), 8 waves,
//         full T loop in-kernel; h exchanged through LDS; weights from L2.

#define Bsz   256
#define Tlen  1024
#define Idim  64
#define Hdim  256
#define Odim  64
#define BTILE 16
#define EPSV  1e-8f
#define HPAD  260   // 256 + 4 pad -> conflict-free column access
#define XPAD  68    // 64 + 4 pad

// float offsets of packed planes inside d_ws
#define WS_RECRE 0          // 128 rowpairs x 256 n x v2f = 65536 floats
#define WS_RECIM 65536
#define WS_INRE  131072     // 32 rowpairs x 256 n x v2f = 16384 floats
#define WS_INIM  147456
#define WS_OUT   163840     // 256 rowpairs x 64 n x v2f = 32768 floats
#define WS_TOTAL 196608     // floats (768 KB)

typedef __attribute__((ext_vector_type(2))) float v2f;
typedef __attribute__((ext_vector_type(4))) float v4f;
typedef __attribute__((ext_vector_type(8))) float v8f;

// D = A(16x4) * B(4x16) + C, fp32. All modifiers immediate-zero
// (f32 WMMA allows only CNeg in NEG[2]; A/B neg must be 0).
__device__ __forceinline__ v8f wmma4(v2f a, v2f b, v8f c) {
  return __builtin_amdgcn_wmma_f32_16x16x4_f32(
      /*neg_a=*/false, a, /*neg_b=*/false, b, /*c_mod=*/(short)0, c,
      /*reuse_a=*/false, /*reuse_b=*/false);
}

// ---------------------------------------------------------------------------
// Pass 1: repack weights. Packed element gid holds v2f {W[2rp][n], W[2rp+1][n]}
// so a lane's B fragment (rows kb,kb+1 / col n) is one b64 load.
// ---------------------------------------------------------------------------
__global__ __launch_bounds__(256) void repack_weights_kernel(
    const float* __restrict__ wrec,  // [H,H] complex64 interleaved
    const float* __restrict__ win,   // [I,H] complex64 interleaved
    const float* __restrict__ wout,  // [2H,O] fp32
    float* __restrict__ ws)
{
  const int gid = blockIdx.x * 256 + threadIdx.x;
  v2f v;
  if (gid < 32768) {                       // W_rec real plane
    int rp = gid >> 8, n = gid & 255;
    v.x = wrec[((size_t)(2 * rp) * Hdim + n) * 2];
    v.y = wrec[((size_t)(2 * rp + 1) * Hdim + n) * 2];
    *(v2f*)(ws + WS_RECRE + (size_t)gid * 2) = v;
  } else if (gid < 65536) {                // W_rec imag plane
    int g = gid - 32768, rp = g >> 8, n = g & 255;
    v.x = wrec[((size_t)(2 * rp) * Hdim + n) * 2 + 1];
    v.y = wrec[((size_t)(2 * rp + 1) * Hdim + n) * 2 + 1];
    *(v2f*)(ws + WS_RECIM + (size_t)g * 2) = v;
  } else if (gid < 73728) {                // W_in real plane
    int g = gid - 65536, rp = g >> 8, n = g & 255;
    v.x = win[((size_t)(2 * rp) * Hdim + n) * 2];
    v.y = win[((size_t)(2 * rp + 1) * Hdim + n) * 2];
    *(v2f*)(ws + WS_INRE + (size_t)g * 2) = v;
  } else if (gid < 81920) {                // W_in imag plane
    int g = gid - 73728, rp = g >> 8, n = g & 255;
    v.x = win[((size_t)(2 * rp) * Hdim + n) * 2 + 1];
    v.y = win[((size_t)(2 * rp + 1) * Hdim + n) * 2 + 1];
    *(v2f*)(ws + WS_INIM + (size_t)g * 2) = v;
  } else if (gid < 98304) {                // W_out
    int g = gid - 81920, rp = g >> 6, n = g & 63;
    v.x = wout[(size_t)(2 * rp) * Odim + n];
    v.y = wout[(size_t)(2 * rp + 1) * Odim + n];
    *(v2f*)(ws + WS_OUT + (size_t)g * 2) = v;
  }
}

// ---------------------------------------------------------------------------
// Pass 2: recurrent kernel.
// ---------------------------------------------------------------------------
__global__ __launch_bounds__(256) void urnn_modrelu_kernel(
    const float* __restrict__ xser,   // [B,T,I] fp32
    const float* __restrict__ wsp,    // packed weight planes
    const float* __restrict__ bin,    // [H] complex64
    const float* __restrict__ bmod,   // [1] fp32
    const float* __restrict__ bout,   // [O] fp32
    const int*   __restrict__ steps_p,
    float* __restrict__ out)          // [B,T,O] fp32
{
  __shared__ float hre[BTILE][HPAD];
  __shared__ float him[BTILE][HPAD];
  __shared__ float xbuf[BTILE][XPAD];

  const float* __restrict__ wRecRe = wsp + WS_RECRE;
  const float* __restrict__ wRecIm = wsp + WS_RECIM;
  const float* __restrict__ wInRe  = wsp + WS_INRE;
  const float* __restrict__ wInIm  = wsp + WS_INIM;
  const float* __restrict__ wOutP  = wsp + WS_OUT;

  const int tid   = threadIdx.x;
  const int lane  = tid & 31;
  const int wave  = tid >> 5;           // 0..7 (wave32)
  const int lc    = lane & 15;          // column within a 16-wide tile
  const int khalf = lane >> 4;          // K sub-offset (A/B frag layout)
  const int b0    = blockIdx.x * BTILE; // batch-row base for this WG
  const int steps = steps_p[0];

  // ---- init: h = 0, stage x_0 into LDS ----
  for (int i = tid; i < BTILE * HPAD; i += 256) {
    (&hre[0][0])[i] = 0.0f;
    (&him[0][0])[i] = 0.0f;
  }
  {
    int m  = tid >> 4;            // 0..15
    int c0 = (tid & 15) * 4;      // 0..60
    v4f v = *(const v4f*)(xser + ((size_t)(b0 + m) * Tlen) * Idim + c0);
    xbuf[m][c0 + 0] = v.x; xbuf[m][c0 + 1] = v.y;
    xbuf[m][c0 + 2] = v.z; xbuf[m][c0 + 3] = v.w;
  }
  __syncthreads();

  const float br = bmod[0];
  const int n0 = wave * 32;       // this wave owns hidden cols [n0, n0+32)

  // complex input bias for our two column tiles (zeros in reference; general)
  const float bre0 = bin[2 * (n0 + lc) + 0];
  const float bim0 = bin[2 * (n0 + lc) + 1];
  const float bre1 = bin[2 * (n0 + 16 + lc) + 0];
  const float bim1 = bin[2 * (n0 + 16 + lc) + 1];

  for (int t = 0; t < steps; ++t) {
    v8f zre0 = {}, zim0 = {}, zre1 = {}, zim1 = {};

    // ---------- z += x_t @ W_in  (x real, W_in complex) ----------
#pragma unroll 4
    for (int k = 0; k < Idim; k += 4) {
      const int kb = k + 2 * khalf;
      const int rp = (k >> 1) + khalf;               // rowpair index
      v2f a; a.x = xbuf[lc][kb]; a.y = xbuf[lc][kb + 1];
      const size_t o = ((size_t)rp * Hdim + n0 + lc) * 2;
      v2f brf0 = *(const v2f*)(wInRe + o);
      v2f brf1 = *(const v2f*)(wInRe + o + 32);
      v2f bif0 = *(const v2f*)(wInIm + o);
      v2f bif1 = *(const v2f*)(wInIm + o + 32);
      zre0 = wmma4(a, brf0, zre0);
      zim0 = wmma4(a, bif0, zim0);
      zre1 = wmma4(a, brf1, zre1);
      zim1 = wmma4(a, bif1, zim1);
    }

    // ---------- z += h @ W_rec  (complex x complex) ----------
#pragma unroll 2
    for (int k = 0; k < Hdim; k += 4) {
      const int kb = k + 2 * khalf;
      const int rp = (k >> 1) + khalf;
      v2f ar; ar.x = hre[lc][kb]; ar.y = hre[lc][kb + 1];
      v2f ai; ai.x = him[lc][kb]; ai.y = him[lc][kb + 1];
      v2f nai; nai.x = -ai.x; nai.y = -ai.y;   // for the z_re subtract
      const size_t o = ((size_t)rp * Hdim + n0 + lc) * 2;
      v2f brf0 = *(const v2f*)(wRecRe + o);
      v2f brf1 = *(const v2f*)(wRecRe + o + 32);
      v2f bif0 = *(const v2f*)(wRecIm + o);
      v2f bif1 = *(const v2f*)(wRecIm + o + 32);
      // z_re += hr*Wr + (-hi)*Wi ; z_im += hr*Wi + hi*Wr
      zre0 = wmma4(ar,  brf0, zre0);
      zre0 = wmma4(nai, bif0, zre0);
      zim0 = wmma4(ar,  bif0, zim0);
      zim0 = wmma4(ai,  brf0, zim0);
      zre1 = wmma4(ar,  brf1, zre1);
      zre1 = wmma4(nai, bif1, zre1);
      zim1 = wmma4(ar,  bif1, zim1);
      zim1 = wmma4(ai,  brf1, zim1);
    }

    // ---------- bias + modReLU:  h = relu(|z|+b)/(|z|+eps) * z ----------
#pragma unroll
    for (int j = 0; j < 8; ++j) {
      float zr = zre0[j] + bre0, zi = zim0[j] + bim0;
      float mod = __builtin_sqrtf(zr * zr + zi * zi);
      float s = fmaxf(mod + br, 0.0f) / (mod + EPSV);
      zre0[j] = s * zr; zim0[j] = s * zi;
      zr = zre1[j] + bre1; zi = zim1[j] + bim1;
      mod = __builtin_sqrtf(zr * zr + zi * zi);
      s = fmaxf(mod + br, 0.0f) / (mod + EPSV);
      zre1[j] = s * zr; zim1[j] = s * zi;
    }

    __syncthreads();  // all reads of h(t-1) and x_t complete

    // write h(t) back to LDS (C/D layout: VGPR j -> row j + 8*khalf)
#pragma unroll
    for (int j = 0; j < 8; ++j) {
      const int m = j + 8 * khalf;
      hre[m][n0 + lc]      = zre0[j];
      him[m][n0 + lc]      = zim0[j];
      hre[m][n0 + 16 + lc] = zre1[j];
      him[m][n0 + 16 + lc] = zim1[j];
    }
    __syncthreads();  // h(t) visible to all waves

    if (wave < 4) {
      // ---------- out_t = [Re h | Im h] @ W_out + b_out ----------
      const int on0 = wave * 16;  // O-tile
      v8f oacc = {};
#pragma unroll 4
      for (int k = 0; k < 2 * Hdim; k += 4) {
        const int kb = k + 2 * khalf;
        const int rp = (k >> 1) + khalf;
        const bool realHalf = (k < Hdim);   // uniform per chunk
        const int kk = realHalf ? kb : kb - Hdim;
        v2f a;
        a.x = realHalf ? hre[lc][kk] : him[lc][kk];
        a.y = realHalf ? hre[lc][kk + 1] : him[lc][kk + 1];
        v2f bw = *(const v2f*)(wOutP + ((size_t)rp * Odim + on0 + lc) * 2);
        oacc = wmma4(a, bw, oacc);
      }
      const float ob = bout[on0 + lc];
      float* op = out + (size_t)b0 * Tlen * Odim + (size_t)t * Odim + on0 + lc;
#pragma unroll
      for (int j = 0; j < 8; ++j) {
        const int m = j + 8 * khalf;
        op[(size_t)m * Tlen * Odim] = oacc[j] + ob;
      }
    } else if (t + 1 < steps) {
      // ---------- waves 4-7: prefetch x_{t+1} into LDS ----------
      const int lid = (wave - 4) * 32 + lane;  // 0..127
      const int m  = lid >> 3;                 // 0..15
      const int c0 = (lid & 7) * 8;            // 0..56
      const float* src =
          xser + ((size_t)(b0 + m) * Tlen + (t + 1)) * Idim + c0;
      v4f v0 = *(const v4f*)src;
      v4f v1 = *(const v4f*)(src + 4);
      xbuf[m][c0 + 0] = v0.x; xbuf[m][c0 + 1] = v0.y;
      xbuf[m][c0 + 2] = v0.z; xbuf[m][c0 + 3] = v0.w;
      xbuf[m][c0 + 4] = v1.x; xbuf[m][c0 + 5] = v1.y;
      xbuf[m][c0 + 6] = v1.z; xbuf[m][c0 + 7] = v1.w;
    }
    __syncthreads();  // out-proj reads done; x_{t+1} staged
  }
}

extern "C" void kernel_launch(void* const* d_in, const int* in_sizes, int n_in,
                              void* d_out, int out_size, void* d_ws, size_t ws_size,
                              hipStream_t stream) {
  const float* xser  = (const float*)d_in[0];  // input_series [B,T,I]
  const float* wrec  = (const float*)d_in[1];  // weight_recurrent complex64
  const float* win   = (const float*)d_in[2];  // weight_input complex64
  const float* bin   = (const float*)d_in[3];  // bias_input complex64
  const float* bmod  = (const float*)d_in[4];  // b (modReLU radius)
  const float* wout  = (const float*)d_in[5];  // weight_output [2H,O]
  const float* bout  = (const float*)d_in[6];  // bias_output [O]
  const int*   steps = (const int*)d_in[7];    // T
  float* outp = (float*)d_out;
  float* wsp  = (float*)d_ws;

  // Pass 1: fragment-major weight repack (98304 v2f elements)
  repack_weights_kernel<<<dim3(98304 / 256), dim3(256), 0, stream>>>(
      wrec, win, wout, wsp);

  // Pass 2: persistent recurrent kernel, 16 batch tiles x 8 wave32 waves
  urnn_modrelu_kernel<<<dim3(Bsz / BTILE), dim3(256), 0, stream>>>(
      xser, wsp, bin, bmod, bout, steps, outp);
}